// gcn_48344151884423
// MI455X (gfx1250) — compile-verified
//
#include <hip/hip_runtime.h>

#define NN 100000
#define EE 1600000
#define F_IN 256
#define F_HID 64
#define F_OUT 4
#define MTILES (NN / 16)          // 6250

typedef float v2f __attribute__((ext_vector_type(2)));
typedef float v8f __attribute__((ext_vector_type(8)));

// ---------------- zero scratch ----------------
__global__ void gcn_zero_kernel(float* __restrict__ p, int n) {
    int i = blockIdx.x * blockDim.x + threadIdx.x;
    if (i < n) p[i] = 0.0f;
}

// ---------------- structural degrees ----------------
__global__ void gcn_degree_kernel(const int* __restrict__ src, const int* __restrict__ dst,
                                  unsigned* __restrict__ outdeg, unsigned* __restrict__ indeg) {
    int e = blockIdx.x * blockDim.x + threadIdx.x;
    if (e >= EE) return;
    atomicAdd(&outdeg[src[e]], 1u);
    atomicAdd(&indeg[dst[e]], 1u);
}

// ---------------- pre-swizzle W1 into WMMA B-fragment order ----------------
// Bsw[kk][nt][lane] = float2 { W1[k0][col], W1[k0+1][col] }
//   k0 = kk*4 + (lane<16 ? 0 : 2), col = nt*16 + lane%16
__global__ void gcn_bswz_kernel(const float* __restrict__ W1, float* __restrict__ Bsw) {
    int i = blockIdx.x * blockDim.x + threadIdx.x;          // 64*4*32 = 8192
    if (i >= 64 * 4 * 32) return;
    int lane = i & 31;
    int nt   = (i >> 5) & 3;
    int kk   = i >> 7;
    int k0   = kk * 4 + ((lane >> 4) << 1);
    int col  = nt * 16 + (lane & 15);
    Bsw[2 * i + 0] = W1[(k0 + 0) * F_HID + col];
    Bsw[2 * i + 1] = W1[(k0 + 1) * F_HID + col];
}

// ---------------- feat = (inputs * rsqrt(outdeg)) @ W1 via V_WMMA_F32_16X16X4_F32 ----
// One wave computes a 16-row x 64-col tile: 4 accumulators, 64 K-steps of 4.
__global__ void gcn_gemm1_kernel(const float* __restrict__ inputs,
                                 const float* __restrict__ Bsw,
                                 const unsigned* __restrict__ outdeg,
                                 float* __restrict__ feat) {
    int wid  = (blockIdx.x * blockDim.x + threadIdx.x) >> 5;
    int lane = threadIdx.x & 31;
    if (wid >= MTILES) return;

    int m0  = wid << 4;
    int row = m0 + (lane & 15);
    unsigned dg = outdeg[row];
    float scale = rsqrtf((float)(dg < 1u ? 1u : dg));

    // A fragment per K-step: lane<16 -> K offset 0, lane>=16 -> K offset 2 (contiguous float2)
    const float2* arow = reinterpret_cast<const float2*>(
        inputs + (size_t)row * F_IN + ((lane >> 4) << 1));
    const v2f* b2 = reinterpret_cast<const v2f*>(Bsw);

    v8f c0 = {}, c1 = {}, c2 = {}, c3 = {};

#pragma unroll 8
    for (int kk = 0; kk < 64; ++kk) {
        float2 av = arow[kk * 2];                 // floats at kk*4 + {0|2}
        v2f a;
        a[0] = av.x * scale;
        a[1] = av.y * scale;
        int bi = (kk * 4) * 32 + lane;
        c0 = __builtin_amdgcn_wmma_f32_16x16x4_f32(false, a, false, b2[bi +  0], (short)0, c0, false, false);
        c1 = __builtin_amdgcn_wmma_f32_16x16x4_f32(false, a, false, b2[bi + 32], (short)0, c1, false, false);
        c2 = __builtin_amdgcn_wmma_f32_16x16x4_f32(false, a, false, b2[bi + 64], (short)0, c2, false, false);
        c3 = __builtin_amdgcn_wmma_f32_16x16x4_f32(false, a, false, b2[bi + 96], (short)0, c3, false, false);
    }

    // C layout: element r, lane n -> M = r + (n<16?0:8), N = n%16. Stores coalesce across lanes.
    int col   = lane & 15;
    int rbase = m0 + ((lane >> 4) << 3);
#pragma unroll
    for (int r = 0; r < 8; ++r) {
        float* fr = feat + (size_t)(rbase + r) * F_HID + col;
        fr[0]  = c0[r];
        fr[16] = c1[r];
        fr[32] = c2[r];
        fr[48] = c3[r];
    }
}

// ---------------- edge gather * w -> scatter-add (16 lanes per edge, float4 each) ----
__global__ void gcn_edge_kernel(const int* __restrict__ src, const int* __restrict__ dst,
                                const float* __restrict__ w, const float* __restrict__ feat,
                                float* __restrict__ agg) {
    int tid = blockIdx.x * blockDim.x + threadIdx.x;
    int e   = tid >> 4;
    int sub = tid & 15;
    if (e >= EE) return;
    int s = src[e];
    int d = dst[e];
    float ww = w[e];
    float4 v = reinterpret_cast<const float4*>(feat + (size_t)s * F_HID)[sub];
    float* ap = agg + (size_t)d * F_HID + sub * 4;
    atomicAdd(ap + 0, v.x * ww);
    atomicAdd(ap + 1, v.y * ww);
    atomicAdd(ap + 2, v.z * ww);
    atomicAdd(ap + 3, v.w * ww);
}

// ---------------- out = relu(agg*rsqrt(indeg)+b1) @ fcW + fcb (16 lanes per node) ----
__global__ void gcn_final_kernel(const float* __restrict__ agg, const unsigned* __restrict__ indeg,
                                 const float* __restrict__ b1, const float* __restrict__ fcW,
                                 const float* __restrict__ fcb, float* __restrict__ out) {
    int tid  = blockIdx.x * blockDim.x + threadIdx.x;
    int node = tid >> 4;
    int sub  = tid & 15;
    if (node >= NN) return;

    unsigned dg = indeg[node];
    float s = rsqrtf((float)(dg < 1u ? 1u : dg));

    float4 a  = reinterpret_cast<const float4*>(agg + (size_t)node * F_HID)[sub];
    float4 bb = reinterpret_cast<const float4*>(b1)[sub];
    float h0 = fmaxf(a.x * s + bb.x, 0.0f);
    float h1 = fmaxf(a.y * s + bb.y, 0.0f);
    float h2 = fmaxf(a.z * s + bb.z, 0.0f);
    float h3 = fmaxf(a.w * s + bb.w, 0.0f);

    const float4* wr = reinterpret_cast<const float4*>(fcW) + sub * 4;  // rows k=sub*4..sub*4+3
    float4 w0 = wr[0], w1 = wr[1], w2 = wr[2], w3 = wr[3];
    float px = w0.x * h0 + w1.x * h1 + w2.x * h2 + w3.x * h3;
    float py = w0.y * h0 + w1.y * h1 + w2.y * h2 + w3.y * h3;
    float pz = w0.z * h0 + w1.z * h1 + w2.z * h2 + w3.z * h3;
    float pw = w0.w * h0 + w1.w * h1 + w2.w * h2 + w3.w * h3;

#pragma unroll
    for (int off = 8; off >= 1; off >>= 1) {
        px += __shfl_down(px, off, 16);
        py += __shfl_down(py, off, 16);
        pz += __shfl_down(pz, off, 16);
        pw += __shfl_down(pw, off, 16);
    }

    if (sub == 0) {
        float4 r;
        r.x = px + fcb[0];
        r.y = py + fcb[1];
        r.z = pz + fcb[2];
        r.w = pw + fcb[3];
        reinterpret_cast<float4*>(out)[node] = r;
    }
}

extern "C" void kernel_launch(void* const* d_in, const int* in_sizes, int n_in,
                              void* d_out, int out_size, void* d_ws, size_t ws_size,
                              hipStream_t stream) {
    (void)in_sizes; (void)n_in; (void)out_size; (void)ws_size;
    const float* inputs = (const float*)d_in[0];
    const float* ew     = (const float*)d_in[1];
    const float* W1     = (const float*)d_in[2];
    const float* b1     = (const float*)d_in[3];
    const float* fcW    = (const float*)d_in[4];
    const float* fcb    = (const float*)d_in[5];
    const int*   src    = (const int*)d_in[6];
    const int*   dst    = (const int*)d_in[7];
    float* out = (float*)d_out;

    char* ws = (char*)d_ws;
    // byte offsets, all 128B-aligned
    unsigned* outdeg = (unsigned*)(ws + 0);            //   400,000 B
    unsigned* indeg  = (unsigned*)(ws + 400000);       //   400,000 B
    float*    feat   = (float*)(ws + 800000);          // 25,600,000 B
    float*    agg    = (float*)(ws + 26400000);        // 25,600,000 B
    float*    Bsw    = (float*)(ws + 52000000);        //     65,536 B

    // zero degrees (200,000 u32 contiguous at ws start) and agg
    gcn_zero_kernel<<<(200000 + 255) / 256, 256, 0, stream>>>((float*)ws, 200000);
    gcn_zero_kernel<<<(NN * F_HID + 255) / 256, 256, 0, stream>>>(agg, NN * F_HID);

    gcn_degree_kernel<<<(EE + 255) / 256, 256, 0, stream>>>(src, dst, outdeg, indeg);
    gcn_bswz_kernel<<<8192 / 256, 256, 0, stream>>>(W1, Bsw);
    gcn_gemm1_kernel<<<(MTILES * 32 + 255) / 256, 256, 0, stream>>>(inputs, Bsw, outdeg, feat);
    gcn_edge_kernel<<<(EE * 16) / 256, 256, 0, stream>>>(src, dst, ew, feat, agg);
    gcn_final_kernel<<<(NN * 16) / 256, 256, 0, stream>>>(agg, indeg, b1, fcW, fcb, out);
}